// myHeteroGATConv_32435593019977
// MI455X (gfx1250) — compile-verified
//
#include <hip/hip_runtime.h>
#include <hip/hip_bf16.h>
#include <math.h>

typedef __attribute__((ext_vector_type(2))) float v2f;
typedef __attribute__((ext_vector_type(8))) float v8f;

#define IN_FEATS   128
#define HEADS      4
#define OUT_FEATS  16
#define HD         64      // HEADS*OUT_FEATS
#define EDGE_FEATS 64
#define NEG_SLOPE  0.2f

// ---------------------------------------------------------------------------
// atomic max for float with mixed signs (memory initialized to -inf)
// ---------------------------------------------------------------------------
__device__ __forceinline__ void atomicMaxF(float* addr, float val) {
    if (val >= 0.0f) {
        atomicMax((int*)addr, __float_as_int(val));
    } else {
        atomicMin((unsigned int*)addr, __float_as_uint(val));
    }
}

// ---------------------------------------------------------------------------
// tiny: ee[t,h] = sum_e ( (edge_emb @ W_e)[t, h*64+e] * attn_e[h,e] )
// ---------------------------------------------------------------------------
__global__ void ee_kernel(const float* __restrict__ edge_emb,
                          const float* __restrict__ W_e,
                          const float* __restrict__ attn_e,
                          float* __restrict__ ee, int n_etypes) {
    int tid = threadIdx.x;
    if (tid >= n_etypes * HEADS) return;
    int t = tid >> 2, h = tid & 3;
    float s = 0.0f;
    for (int eo = 0; eo < EDGE_FEATS; ++eo) {
        float acc = 0.0f;
        for (int k = 0; k < EDGE_FEATS; ++k)
            acc += edge_emb[t * EDGE_FEATS + k] *
                   W_e[k * (HEADS * EDGE_FEATS) + h * EDGE_FEATS + eo];
        s += acc * attn_e[h * EDGE_FEATS + eo];
    }
    ee[t * HEADS + h] = s;
}

// ---------------------------------------------------------------------------
// init: rst = 0, m = -inf, z = 0
// ---------------------------------------------------------------------------
__global__ void init_kernel(float* __restrict__ rst, float* __restrict__ m,
                            float* __restrict__ z, int N) {
    int tid = blockIdx.x * blockDim.x + threadIdx.x;
    if (tid < N * HD)    rst[tid] = 0.0f;
    if (tid < N * HEADS) {
        m[tid] = __int_as_float(0xFF800000);  // -inf
        z[tid] = 0.0f;
    }
}

// ---------------------------------------------------------------------------
// feat = x @ W via V_WMMA_F32_16X16X4_F32, fused with el/er reductions.
// One wave = 16 rows x 64 cols (4 n-tiles; each n-tile == one head).
// ---------------------------------------------------------------------------
__global__ void __launch_bounds__(256)
gemm_feat_kernel(const float* __restrict__ x, const float* __restrict__ W,
                 const float* __restrict__ attn_l, const float* __restrict__ attn_r,
                 float* __restrict__ feat, float* __restrict__ el,
                 float* __restrict__ er, int N) {
    const int lane = threadIdx.x & 31;
    const int wave = threadIdx.x >> 5;
    const int row0 = (blockIdx.x * 8 + wave) * 16;
    if (row0 >= N) return;                 // wave-uniform exit; no barriers used

    const int half = lane >> 4;            // 0: lanes 0-15, 1: lanes 16-31
    const int l16  = lane & 15;

    v8f acc0 = {}; v8f acc1 = {}; v8f acc2 = {}; v8f acc3 = {};

    const int arow = row0 + l16;
    const float* xrow = x + (size_t)arow * IN_FEATS;

    for (int k0 = 0; k0 < IN_FEATS; k0 += 4) {
        const int ka = k0 + half * 2;      // A layout: lanes0-15 K={0,1}, lanes16-31 K={2,3}
        v2f a;
        a.x = xrow[ka];
        a.y = xrow[ka + 1];

        const float* Wk0 = W + (size_t)ka * HD;
        const float* Wk1 = Wk0 + HD;
#pragma unroll
        for (int nt = 0; nt < 4; ++nt) {
            const int col = nt * 16 + l16;
            v2f b;
            b.x = Wk0[col];
            b.y = Wk1[col];
            v8f* accp = (nt == 0) ? &acc0 : (nt == 1) ? &acc1 : (nt == 2) ? &acc2 : &acc3;
            *accp = __builtin_amdgcn_wmma_f32_16x16x4_f32(
                false, a, false, b, (short)0, *accp, false, false);
        }
    }

    // D layout: VGPR r holds (row0+r, col) for lanes 0-15 and (row0+r+8, col)
    // for lanes 16-31, col = nt*16 + l16.  n-tile nt == head nt; d == l16.
    v8f accs[4] = {acc0, acc1, acc2, acc3};
#pragma unroll
    for (int nt = 0; nt < 4; ++nt) {
        const float aL = attn_l[nt * OUT_FEATS + l16];
        const float aR = attn_r[nt * OUT_FEATS + l16];
#pragma unroll
        for (int r = 0; r < 8; ++r) {
            const int row = row0 + r + half * 8;
            const float v = accs[nt][r];
            feat[(size_t)row * HD + nt * 16 + l16] = v;
            // reduce v*attn over the 16 lanes of this half-wave
            float pL = v * aL, pR = v * aR;
#pragma unroll
            for (int off = 8; off >= 1; off >>= 1) {
                pL += __shfl_xor(pL, off, 32);
                pR += __shfl_xor(pR, off, 32);
            }
            if (l16 == 0) {
                el[row * HEADS + nt] = pL;
                er[row * HEADS + nt] = pR;
            }
        }
    }
}

// ---------------------------------------------------------------------------
// pass 1: s = leaky(el[src]+er[dst]+ee[etype]); segment max into m
// ---------------------------------------------------------------------------
__global__ void score_kernel(const float* __restrict__ el, const float* __restrict__ er,
                             const float* __restrict__ ee,
                             const int* __restrict__ src, const int* __restrict__ dst,
                             const int* __restrict__ ety,
                             float* __restrict__ s, float* __restrict__ m, int E) {
    int tid = blockIdx.x * blockDim.x + threadIdx.x;
    if (tid >= E * HEADS) return;
    const int e = tid >> 2, h = tid & 3;
    const int d = dst[e];
    float v = el[src[e] * HEADS + h] + er[d * HEADS + h] + ee[ety[e] * HEADS + h];
    v = (v > 0.0f) ? v : NEG_SLOPE * v;
    s[tid] = v;
    atomicMaxF(&m[d * HEADS + h], v);
}

// ---------------------------------------------------------------------------
// pass 2: e = exp(s - m[dst]) -> out_a; segment sum into z
// ---------------------------------------------------------------------------
__global__ void expsum_kernel(const float* __restrict__ s, const float* __restrict__ m,
                              const int* __restrict__ dst,
                              float* __restrict__ out_a, float* __restrict__ z, int E) {
    int tid = blockIdx.x * blockDim.x + threadIdx.x;
    if (tid >= E * HEADS) return;
    const int e = tid >> 2, h = tid & 3;
    const int d = dst[e];
    const float v = __expf(s[tid] - m[d * HEADS + h]);
    out_a[tid] = v;
    unsafeAtomicAdd(&z[d * HEADS + h], v);
}

// ---------------------------------------------------------------------------
// pass 3a: out_a /= z[dst]
// ---------------------------------------------------------------------------
__global__ void norm_kernel(float* __restrict__ out_a, const float* __restrict__ z,
                            const int* __restrict__ dst, int E) {
    int tid = blockIdx.x * blockDim.x + threadIdx.x;
    if (tid >= E * HEADS) return;
    const int e = tid >> 2, h = tid & 3;
    out_a[tid] = out_a[tid] / z[dst[e] * HEADS + h];
}

// ---------------------------------------------------------------------------
// pass 3b: rst[dst] += feat[src] * a   (64 threads per edge, float atomics)
// ---------------------------------------------------------------------------
__global__ void aggregate_kernel(const float* __restrict__ feat, const float* __restrict__ out_a,
                                 const int* __restrict__ src, const int* __restrict__ dst,
                                 float* __restrict__ rst, int E) {
    int tid = blockIdx.x * blockDim.x + threadIdx.x;
    if (tid >= E * HD) return;
    const int e = tid >> 6;
    const int c = tid & 63;      // c = h*16 + d
    const int h = c >> 4;
    const float v = feat[(size_t)src[e] * HD + c] * out_a[e * HEADS + h];
    unsafeAtomicAdd(&rst[(size_t)dst[e] * HD + c], v);
}

// ---------------------------------------------------------------------------
extern "C" void kernel_launch(void* const* d_in, const int* in_sizes, int n_in,
                              void* d_out, int out_size, void* d_ws, size_t ws_size,
                              hipStream_t stream) {
    const float* x        = (const float*)d_in[0];
    const float* W        = (const float*)d_in[1];
    const float* W_e      = (const float*)d_in[2];
    const float* attn_l   = (const float*)d_in[3];
    const float* attn_r   = (const float*)d_in[4];
    const float* attn_e   = (const float*)d_in[5];
    const float* edge_emb = (const float*)d_in[6];
    const int*   src      = (const int*)d_in[7];
    const int*   dst      = (const int*)d_in[8];
    const int*   ety      = (const int*)d_in[9];

    const int N        = in_sizes[0] / IN_FEATS;
    const int E        = in_sizes[7];
    const int n_etypes = in_sizes[6] / EDGE_FEATS;

    float* out_rst = (float*)d_out;                    // [N, H, D]
    float* out_a   = out_rst + (size_t)N * HD;         // [E, H]

    // workspace layout (floats)
    float* feat = (float*)d_ws;                        // N*64
    float* el   = feat + (size_t)N * HD;               // N*4
    float* er   = el   + (size_t)N * HEADS;            // N*4
    float* m    = er   + (size_t)N * HEADS;            // N*4
    float* z    = m    + (size_t)N * HEADS;            // N*4
    float* s    = z    + (size_t)N * HEADS;            // E*4
    float* ee   = s    + (size_t)E * HEADS;            // n_etypes*4

    const int TB = 256;

    ee_kernel<<<1, 64, 0, stream>>>(edge_emb, W_e, attn_e, ee, n_etypes);

    init_kernel<<<(N * HD + TB - 1) / TB, TB, 0, stream>>>(out_rst, m, z, N);

    const int rowTiles = (N + 15) / 16;                // 16 rows per wave
    gemm_feat_kernel<<<(rowTiles + 7) / 8, TB, 0, stream>>>(
        x, W, attn_l, attn_r, feat, el, er, N);

    const int eh = E * HEADS;
    score_kernel<<<(eh + TB - 1) / TB, TB, 0, stream>>>(el, er, ee, src, dst, ety, s, m, E);

    expsum_kernel<<<(eh + TB - 1) / TB, TB, 0, stream>>>(s, m, dst, out_a, z, E);

    norm_kernel<<<(eh + TB - 1) / TB, TB, 0, stream>>>(out_a, z, dst, E);

    const int ec = E * HD;
    aggregate_kernel<<<(ec + TB - 1) / TB, TB, 0, stream>>>(feat, out_a, src, dst, out_rst, E);
}